// KVCacheQuantizer_17927193493786
// MI455X (gfx1250) — compile-verified
//
#include <hip/hip_runtime.h>
#include <stdint.h>

typedef float v4f __attribute__((ext_vector_type(4)));

#define GS 32
#define TPB 256          // 8 waves of 32 on CDNA5
#define EPT 8            // elements per thread (two b128)
// 4 lanes per quantization group (32 elems / 8 per lane)

// ds_swizzle_b32 group-of-32 patterns: offset = xor_mask<<10 | or_mask<<5 | and_mask
#define SWZ_XOR1 0x041F  // lane ^= 1
#define SWZ_XOR2 0x081F  // lane ^= 2

// |x| has sign bit clear -> IEEE order == u32 order: butterfly max on raw bits,
// no fp canonicalization instructions needed.
#define SWZ_UMAX(mb, pat)                                                      \
    do {                                                                       \
        uint32_t _o = (uint32_t)__builtin_amdgcn_ds_swizzle((int)(mb), (pat)); \
        (mb) = (mb) > _o ? (mb) : _o;                                          \
    } while (0)

__global__ __launch_bounds__(TPB) void kvq_int4_roundtrip(
    const float* __restrict__ in,     // [n] source (k or v), group = 32 contiguous
    float*       __restrict__ q_out,  // [n] quantized values stored as f32
    float*       __restrict__ s_out,  // [n/32] per-group scales
    float*       __restrict__ dq_out) // [n] dequantized values
{
    // Two float4 per thread staged through LDS via the gfx1250 async copy engine.
    __shared__ v4f stage[TPB * 2];

    const uint32_t t   = threadIdx.x;
    const uint64_t tid = (uint64_t)blockIdx.x * TPB + t;     // 8-elem chunk index
    const uint64_t e0  = tid * EPT;                          // element index

    // ---- async global -> LDS (2 x b128), tracked with ASYNCcnt ----
    const uint64_t gaddr  = (uint64_t)(uintptr_t)(in + e0);
    const uint32_t laddr0 = (uint32_t)(uintptr_t)&stage[2 * t];
    const uint32_t laddr1 = laddr0 + 16;
    asm volatile("global_load_async_to_lds_b128 %0, %2, off\n\t"
                 "global_load_async_to_lds_b128 %1, %2, off offset:16"
                 :: "v"(laddr0), "v"(laddr1), "v"(gaddr) : "memory");
    asm volatile("s_wait_asynccnt 0x0" ::: "memory");

    const v4f x0 = stage[2 * t];                             // ds_load_b128
    const v4f x1 = stage[2 * t + 1];                         // ds_load_b128

    // ---- group abs-max: local 8-way (max3 trees, free |.| modifiers),
    //      then xor-butterfly over the 4-lane subgroup on raw bits ----
    float m = fmaxf(fmaxf(fmaxf(fabsf(x0.x), fabsf(x0.y)),
                          fmaxf(fabsf(x0.z), fabsf(x0.w))),
                    fmaxf(fmaxf(fabsf(x1.x), fabsf(x1.y)),
                          fmaxf(fabsf(x1.z), fabsf(x1.w))));
    uint32_t mb = __float_as_uint(m);
    SWZ_UMAX(mb, SWZ_XOR1);
    SWZ_UMAX(mb, SWZ_XOR2);
    m = __uint_as_float(mb);

    const float scale = fmaxf(m, 1e-8f) * (1.0f / 7.0f);
    const float inv_s = __builtin_amdgcn_rcpf(scale);        // v_rcp_f32, 1 ulp

    // ---- quantize (round-half-even like jnp.round), clamp to [-8, 7] ----
    v4f q0, q1, dq0, dq1;
    #pragma unroll
    for (int i = 0; i < 4; ++i) {
        float qf = fminf(fmaxf(rintf(x0[i] * inv_s), -8.0f), 7.0f); // v_med3
        q0[i]  = qf;
        dq0[i] = qf * scale;
    }
    #pragma unroll
    for (int i = 0; i < 4; ++i) {
        float qf = fminf(fmaxf(rintf(x1[i] * inv_s), -8.0f), 7.0f);
        q1[i]  = qf;
        dq1[i] = qf * scale;
    }

    // ---- streaming (non-temporal) b128 stores; output is never re-read ----
    __builtin_nontemporal_store(q0,  (v4f*)q_out  + 2 * tid);
    __builtin_nontemporal_store(q1,  (v4f*)q_out  + 2 * tid + 1);
    __builtin_nontemporal_store(dq0, (v4f*)dq_out + 2 * tid);
    __builtin_nontemporal_store(dq1, (v4f*)dq_out + 2 * tid + 1);
    if ((t & 3u) == 0u)                                      // lane 0 of each group
        __builtin_nontemporal_store(scale, s_out + (tid >> 2));
}

extern "C" void kernel_launch(void* const* d_in, const int* in_sizes, int n_in,
                              void* d_out, int out_size, void* d_ws, size_t ws_size,
                              hipStream_t stream) {
    (void)n_in; (void)out_size; (void)d_ws; (void)ws_size;

    const float* k = (const float*)d_in[0];
    const float* v = (const float*)d_in[1];

    const long long N = (long long)in_sizes[0];   // 4*32*4096*128 = 67,108,864
    const long long G = N / GS;                   // 2,097,152 groups per tensor

    // Output order: k_q [N], k_scale [G], v_q [N], v_scale [G], k_dq [N], v_dq [N]
    float* out  = (float*)d_out;
    float* k_q  = out;
    float* k_s  = k_q + N;
    float* v_q  = k_s + G;
    float* v_s  = v_q + N;
    float* k_dq = v_s + G;
    float* v_dq = k_dq + N;

    const int threads = TPB;
    const int blocks  = (int)(N / (TPB * EPT));   // 32,768 — exact, no tail

    kvq_int4_roundtrip<<<blocks, threads, 0, stream>>>(k, k_q, k_s, k_dq);
    kvq_int4_roundtrip<<<blocks, threads, 0, stream>>>(v, v_q, v_s, v_dq);
}